// StemBlock_86887188398529
// MI455X (gfx1250) — compile-verified
//
#include <hip/hip_runtime.h>

// StemBlock for MI455X (gfx1250, wave32).
// Heavy ops (conv1: 11 GFLOP, conv2: 22 GFLOP, point linear: 0.8 GFLOP) run on
// V_WMMA_F32_16X16X4_F32 (native fp32 WMMA). Working set is L2-resident
// (192 MB L2 >> 25.6 MB per feature buffer), so gathers are cheap and the
// kernel is fp32-matrix-compute bound -> WMMA f32 is the right path.
// Conv waves block 2 M-tiles (32 rows x 64 cols, 8x v8f accumulators) so each
// B fragment feeds 2 WMMAs (~6 B loaded per WMMA instead of ~10).

#define NI 32
#define NF 64
#define KK 27
#define II 8
#define BN_EPS 1e-5f

typedef float v2f __attribute__((ext_vector_type(2)));
typedef float v8f __attribute__((ext_vector_type(8)));

// silu via single v_rcp_f32 (avoid the ~10-op IEEE divide expansion)
__device__ __forceinline__ float silu_f(float x) {
    return x * __builtin_amdgcn_rcpf(1.0f + __expf(-x));
}

__device__ __forceinline__ v8f wmma_f32_4(v2f a, v2f b, v8f c) {
    // V_WMMA_F32_16X16X4_F32: D = A(16x4) x B(4x16) + C(16x16)
    return __builtin_amdgcn_wmma_f32_16x16x4_f32(
        /*neg_a=*/false, a, /*neg_b=*/false, b,
        /*c_mod=*/(short)0, c, /*reuse_a=*/false, /*reuse_b=*/false);
}

// ---------------------------------------------------------------------------
// Grouped conv: out[n, o] = sum_k sum_i in[nbr[n,k], i] * w[k, i, o]
// One wave per 32-row block (2 WMMA M-tiles), covering all NF=64 output
// columns (4 col-tiles): 8 accumulators; A fragments reused 4x, B fragments 2x.
// A layout (16x4 f32): lane&15 = row M; VGPR pair holds K = 2*(lane>>4)+{0,1}.
// B layout (4x16 f32): lane&15 = col N; VGPR pair holds K = 2*(lane>>4)+{0,1}.
// C/D layout: VGPR r, lanes 0-15 -> (M=r, N=lane); lanes 16-31 -> (M=r+8).
// ---------------------------------------------------------------------------
template <int IC>
__global__ void conv_wmma_kernel(const float* __restrict__ in,
                                 const int* __restrict__ nbr,
                                 const float* __restrict__ w,
                                 float* __restrict__ out, int Nv) {
    const int wave = (int)((blockIdx.x * blockDim.x + threadIdx.x) >> 5);
    const int lane = (int)(threadIdx.x & 31u);
    const int n0 = wave * 32;                   // 2 M-tiles per wave
    if (n0 >= Nv) return;                       // wave-uniform exit

    const int m = lane & 15;
    const int khalf = (lane >> 4) << 1;         // 0 or 2
    int row0 = n0 + m;
    int row1 = n0 + 16 + m;
    if (row0 >= Nv) row0 = Nv - 1;              // clamp reads (no divergence)
    if (row1 >= Nv) row1 = Nv - 1;

    v8f acc[2][4];
    #pragma unroll
    for (int s = 0; s < 2; ++s)
        #pragma unroll
        for (int t = 0; t < 4; ++t)
            #pragma unroll
            for (int e = 0; e < 8; ++e) acc[s][t][e] = 0.0f;

    const int col = lane & 15;
    for (int kk = 0; kk < KK; ++kk) {
        const int nb0 = nbr[row0 * KK + kk];
        const int nb1 = nbr[row1 * KK + kk];
        const float* ar0 = in + (long long)nb0 * IC;
        const float* ar1 = in + (long long)nb1 * IC;
        const float* wk = w + (long long)kk * IC * NF;
        for (int ib = 0; ib < IC; ib += 4) {
            v2f a0, a1;
            a0.x = ar0[ib + khalf];
            a0.y = ar0[ib + khalf + 1];
            a1.x = ar1[ib + khalf];
            a1.y = ar1[ib + khalf + 1];
            const float* wr = wk + (long long)(ib + khalf) * NF;
            #pragma unroll
            for (int t = 0; t < 4; ++t) {
                v2f b;
                b.x = wr[t * 16 + col];
                b.y = wr[NF + t * 16 + col];
                acc[0][t] = wmma_f32_4(a0, b, acc[0][t]);
                acc[1][t] = wmma_f32_4(a1, b, acc[1][t]);
            }
        }
    }

    #pragma unroll
    for (int s = 0; s < 2; ++s) {
        const int rbase = n0 + s * 16 + ((lane >> 4) << 3);
        #pragma unroll
        for (int t = 0; t < 4; ++t) {
            #pragma unroll
            for (int r = 0; r < 8; ++r) {
                const int rr = rbase + r;
                if (rr < Nv) out[(long long)rr * NF + (t * 16 + col)] = acc[s][t][r];
            }
        }
    }
}

// ---------------------------------------------------------------------------
// BatchNorm column statistics: per-channel sum and sum-of-squares over rows.
// blockDim.x = 256; each block reduces locally, then a few global f32 atomics.
// ---------------------------------------------------------------------------
__global__ void bnstats_kernel(const float* __restrict__ v, int rows, int C,
                               float* __restrict__ sum, float* __restrict__ sumsq) {
    const int col = (int)(threadIdx.x) % C;
    const int rg = (int)(threadIdx.x) / C;
    const int groups = (int)blockDim.x / C;
    float s = 0.0f, s2 = 0.0f;
    for (long long r = (long long)blockIdx.x * groups + rg; r < rows;
         r += (long long)gridDim.x * groups) {
        const float x = v[r * C + col];
        s += x;
        s2 += x * x;
    }
    __shared__ float ls[256];
    __shared__ float ls2[256];
    ls[threadIdx.x] = s;
    ls2[threadIdx.x] = s2;
    __syncthreads();
    if (rg == 0) {
        for (int g = 1; g < groups; ++g) {
            s += ls[g * C + col];
            s2 += ls2[g * C + col];
        }
        atomicAdd(&sum[col], s);
        atomicAdd(&sumsq[col], s2);
    }
}

// a[c], b[c] so that BN(x) = a*x + b  (population variance, ddof=0)
__global__ void bnfinal_kernel(const float* __restrict__ sum,
                               const float* __restrict__ sumsq,
                               const float* __restrict__ gamma,
                               const float* __restrict__ beta, float n,
                               float* __restrict__ a, float* __restrict__ b, int C) {
    const int c = (int)threadIdx.x;
    if (c < C) {
        const float mean = sum[c] / n;
        const float var = sumsq[c] / n - mean * mean;
        const float inv = rsqrtf(var + BN_EPS);
        const float aa = gamma[c] * inv;
        a[c] = aa;
        b[c] = beta[c] - mean * aa;
    }
}

// in-place y = silu(a*x + b)
__global__ void bnsilu_kernel(float* __restrict__ v, const float* __restrict__ a,
                              const float* __restrict__ b, long long n, int C) {
    const long long i = (long long)blockIdx.x * blockDim.x + threadIdx.x;
    if (i < n) {
        const int c = (int)(i % C);
        v[i] = silu_f(a[c] * v[i] + b[c]);
    }
}

// ---------------------------------------------------------------------------
// Point branch: z1 = trilinear_gather(x) + silu(BN(z)) @ lin_w + lin_b.
// WMMA over K=32 (8 k-steps); BN+SiLU fused into A-fragment loads; the 8-way
// interpolation gather and bias are folded into the accumulator before store.
// ---------------------------------------------------------------------------
__global__ void point_wmma_kernel(const float* __restrict__ xv,   // [N, NF]
                                  const float* __restrict__ zf,   // [P, NI]
                                  const float* __restrict__ iw,   // [P, II]
                                  const int* __restrict__ iidx,   // [P, II]
                                  const float* __restrict__ linw, // [NI, NF]
                                  const float* __restrict__ linb, // [NF]
                                  const float* __restrict__ ap,   // [NI]
                                  const float* __restrict__ bp,   // [NI]
                                  float* __restrict__ z1, int P) {
    const int wave = (int)((blockIdx.x * blockDim.x + threadIdx.x) >> 5);
    const int lane = (int)(threadIdx.x & 31u);
    const int p0 = wave * 16;
    if (p0 >= P) return;                        // wave-uniform exit

    const int m = lane & 15;
    const int khalf = (lane >> 4) << 1;
    int prow = p0 + m;
    if (prow >= P) prow = P - 1;

    v8f acc[4];
    #pragma unroll
    for (int t = 0; t < 4; ++t)
        #pragma unroll
        for (int e = 0; e < 8; ++e) acc[t][e] = 0.0f;

    const int col = lane & 15;
    const float* zr = zf + (long long)prow * NI;
    for (int kb = 0; kb < NI; kb += 4) {
        const int k0 = kb + khalf;
        v2f a;
        a.x = silu_f(ap[k0] * zr[k0] + bp[k0]);
        a.y = silu_f(ap[k0 + 1] * zr[k0 + 1] + bp[k0 + 1]);
        const float* wr = linw + (long long)k0 * NF;
        #pragma unroll
        for (int t = 0; t < 4; ++t) {
            v2f b;
            b.x = wr[t * 16 + col];
            b.y = wr[NF + t * 16 + col];
            acc[t] = wmma_f32_4(a, b, acc[t]);
        }
    }

    const int rbase = p0 + ((lane >> 4) << 3);
    #pragma unroll
    for (int r = 0; r < 8; ++r) {
        const int pr = rbase + r;
        const bool ok = pr < P;
        const int pc = ok ? pr : P - 1;
        int idx[II];
        float wgt[II];
        #pragma unroll
        for (int i = 0; i < II; ++i) {
            idx[i] = iidx[pc * II + i];
            wgt[i] = iw[pc * II + i];
        }
        #pragma unroll
        for (int t = 0; t < 4; ++t) {
            const int c = t * 16 + col;
            float s = acc[t][r] + linb[c];
            #pragma unroll
            for (int i = 0; i < II; ++i)
                s += xv[(long long)idx[i] * NF + c] * wgt[i];
            if (ok) z1[(long long)pr * NF + c] = s;
        }
    }
}

// scatter-add point features into voxel sums + occupancy counts
__global__ void scatter_kernel(const float* __restrict__ z1,
                               const int* __restrict__ p2v,
                               float* __restrict__ sums, float* __restrict__ cnt,
                               long long total) {
    const long long tid = (long long)blockIdx.x * blockDim.x + threadIdx.x;
    if (tid >= total) return;
    const int p = (int)(tid >> 6);  // / NF
    const int c = (int)(tid & 63);  // % NF
    const int v = p2v[p];
    atomicAdd(&sums[(long long)v * NF + c], z1[tid]);
    if (c == 0) atomicAdd(&cnt[v], 1.0f);
}

__global__ void avg_kernel(const float* __restrict__ sums,
                           const float* __restrict__ cnt,
                           float* __restrict__ out, long long total) {
    const long long tid = (long long)blockIdx.x * blockDim.x + threadIdx.x;
    if (tid >= total) return;
    const int n = (int)(tid >> 6);
    float c = cnt[n];
    c = c > 1.0f ? c : 1.0f;
    out[tid] = sums[tid] / c;
}

extern "C" void kernel_launch(void* const* d_in, const int* in_sizes, int n_in,
                              void* d_out, int out_size, void* d_ws, size_t ws_size,
                              hipStream_t stream) {
    const float* xf   = (const float*)d_in[0];   // [N, NI]
    const float* zf   = (const float*)d_in[1];   // [P, NI]
    const float* iw   = (const float*)d_in[2];   // [P, II]
    const int*   nbr  = (const int*)d_in[3];     // [N, KK]
    const int*   iidx = (const int*)d_in[4];     // [P, II]
    const int*   p2v  = (const int*)d_in[5];     // [P]
    const float* w1   = (const float*)d_in[6];   // [KK, NI, NF]
    const float* w2   = (const float*)d_in[7];   // [KK, NF, NF]
    const float* bng  = (const float*)d_in[8];   // [NF]
    const float* bnb  = (const float*)d_in[9];   // [NF]
    const float* pbng = (const float*)d_in[10];  // [NI]
    const float* pbnb = (const float*)d_in[11];  // [NI]
    const float* linw = (const float*)d_in[12];  // [NI, NF]
    const float* linb = (const float*)d_in[13];  // [NF]

    const int Nv = in_sizes[0] / NI;
    const int P  = in_sizes[5];

    // workspace layout (floats): h-buffer (reused as scatter sums), x-buffer,
    // counts, BN stats. Total ~ (2*N*64 + N + 384) * 4B ~= 52 MB.
    float* W     = (float*)d_ws;
    float* hbuf  = W;                                  // [N, NF]
    float* xbuf  = hbuf + (long long)Nv * NF;          // [N, NF]
    float* cnt   = xbuf + (long long)Nv * NF;          // [N]
    float* st    = cnt + Nv;                           // 384 floats
    float* sum64 = st;
    float* sq64  = st + 64;
    float* a64   = st + 128;
    float* b64   = st + 192;
    float* sum32 = st + 256;
    float* sq32  = st + 288;
    float* a32   = st + 320;
    float* b32   = st + 352;

    float* xnew = (float*)d_out;                        // [N, NF]
    float* z1   = (float*)d_out + (long long)Nv * NF;   // [P, NF]

    hipMemsetAsync(st, 0, 384 * sizeof(float), stream);
    hipMemsetAsync(cnt, 0, (size_t)Nv * sizeof(float), stream);

    const dim3 blk(128);  // 4 waves per block
    const int blocksN = (Nv + 31) / 32;   // one 32-row block per wave
    const int tilesP  = (P + 15) / 16;    // one 16-row tile per wave
    const dim3 grdN((blocksN + 3) / 4);
    const dim3 grdP((tilesP + 3) / 4);
    const long long nh = (long long)Nv * NF;
    const long long np = (long long)P * NF;

    // conv1 -> BN stats -> BN+SiLU (in place) -> conv2
    conv_wmma_kernel<NI><<<grdN, blk, 0, stream>>>(xf, nbr, w1, hbuf, Nv);
    bnstats_kernel<<<256, 256, 0, stream>>>(hbuf, Nv, NF, sum64, sq64);
    bnfinal_kernel<<<1, 64, 0, stream>>>(sum64, sq64, bng, bnb, (float)Nv, a64, b64, 64);
    bnsilu_kernel<<<(unsigned)((nh + 255) / 256), 256, 0, stream>>>(hbuf, a64, b64, nh, NF);
    conv_wmma_kernel<NF><<<grdN, blk, 0, stream>>>(hbuf, nbr, w2, xbuf, Nv);

    // point-branch BN stats
    bnstats_kernel<<<256, 256, 0, stream>>>(zf, P, NI, sum32, sq32);
    bnfinal_kernel<<<1, 32, 0, stream>>>(sum32, sq32, pbng, pbnb, (float)P, a32, b32, 32);

    // reuse hbuf as scatter sums (conv2 finished reading it; stream-ordered)
    hipMemsetAsync(hbuf, 0, (size_t)nh * sizeof(float), stream);

    // trilinear gather + fused BN/SiLU/linear -> z1 (written directly to d_out)
    point_wmma_kernel<<<grdP, blk, 0, stream>>>(xbuf, zf, iw, iidx, linw, linb,
                                                a32, b32, z1, P);

    // point_to_voxel scatter-mean -> x_new
    scatter_kernel<<<(unsigned)((np + 255) / 256), 256, 0, stream>>>(z1, p2v, hbuf, cnt, np);
    avg_kernel<<<(unsigned)((nh + 255) / 256), 256, 0, stream>>>(hbuf, cnt, xnew, nh);
}